// SubpixelReshuffleLayer_11132555231952
// MI455X (gfx1250) — compile-verified
//
#include <hip/hip_runtime.h>
#include <stdint.h>

// Problem constants (fixed by the reference setup_inputs()):
//   x: [32, 256, 256, 12] f32  ->  out: [32, 512, 512, 3] f32
//   out[b, 2i+r2, 2j+r1, g] = x[b, i, j, g*4 + r1*2 + r2]
#define IN_W       256
#define CIN        12                 // C * r * r
#define ROW_F      (IN_W * CIN)       // 3072 floats per input row  (12 KB)
#define OUT_ROW_F  (IN_W * 2 * 3)     // 1536 floats per output row (6 KB)

typedef int   v4i __attribute__((ext_vector_type(4)));
typedef float v4f __attribute__((ext_vector_type(4)));

typedef __attribute__((address_space(1))) v4i GV4;   // global int4
typedef __attribute__((address_space(3))) v4i LV4;   // LDS int4

// ---- CDNA5 async global -> LDS copy (ASYNCcnt path) --------------------
__device__ __forceinline__ void async_copy_b128(const float* g, float* l) {
#if __has_builtin(__builtin_amdgcn_global_load_async_to_lds_b128)
  __builtin_amdgcn_global_load_async_to_lds_b128(
      (GV4*)const_cast<float*>(g), (LV4*)l, /*offset=*/0, /*cpol=*/0);
#else
  uint32_t lds = (uint32_t)(size_t)(__attribute__((address_space(3))) void*)l;
  uint64_t ga  = (uint64_t)(size_t)g;
  asm volatile("global_load_async_to_lds_b128 %0, %1, off"
               :: "v"(lds), "v"(ga) : "memory");
#endif
}

__device__ __forceinline__ void wait_async0() {
#if __has_builtin(__builtin_amdgcn_s_wait_asynccnt)
  __builtin_amdgcn_s_wait_asynccnt(0);
#else
  asm volatile("s_wait_asynccnt 0" ::: "memory");
#endif
}

// ------------------------------------------------------------------------
// One workgroup per input row (b,i): 256 threads, one input pixel each.
//  Stage 1: async-copy the raw 12KB input row into LDS (b128, ASYNCcnt).
//  Stage 2: each lane permutes its own 12 floats into the two output-row
//           LDS buffers (each lane only touches data its own wave copied,
//           so s_wait_asynccnt 0 alone orders stage 1 -> stage 2).
//  Stage 3: barrier, then cooperative 16B-aligned coalesced b128 NT stores
//           of the two 6KB output rows.
// ------------------------------------------------------------------------
__global__ __launch_bounds__(256)
void subpixel_reshuffle_kernel(const float* __restrict__ x,
                               float* __restrict__ out) {
  __shared__ float sin_raw[ROW_F];          // 12 KB: raw input row
  __shared__ float sout[2 * OUT_ROW_F];     // 12 KB: two permuted output rows

  const int row = blockIdx.x;               // row = b*256 + i  (0..8191)
  const int j   = threadIdx.x;              // input column     (0..255)

  // ---- Stage 1: async global -> LDS (3 x b128 = 48B per lane) ----
  const float* g  = x + (size_t)row * ROW_F + (size_t)j * CIN;
  float*       li = &sin_raw[j * CIN];
  async_copy_b128(g,     li);
  async_copy_b128(g + 4, li + 4);
  async_copy_b128(g + 8, li + 8);
  wait_async0();

  // ---- Stage 2: per-lane permutation (registers), write to out LDS ----
  // v[idx], idx = g*4 + r1*2 + r2
  v4f a = *(const v4f*)(li);                // v0..v3   (ds_load_b128)
  v4f b = *(const v4f*)(li + 4);            // v4..v7
  v4f c = *(const v4f*)(li + 8);            // v8..v11

  // output row 2*row   (r2=0): pixels (2j): v0,v4,v8   (2j+1): v2,v6,v10
  float* o0 = &sout[6 * j];
  o0[0] = a.x; o0[1] = b.x; o0[2] = c.x;
  o0[3] = a.z; o0[4] = b.z; o0[5] = c.z;
  // output row 2*row+1 (r2=1): pixels (2j): v1,v5,v9   (2j+1): v3,v7,v11
  float* o1 = &sout[OUT_ROW_F + 6 * j];
  o1[0] = a.y; o1[1] = b.y; o1[2] = c.y;
  o1[3] = a.w; o1[4] = b.w; o1[5] = c.w;

  __syncthreads();

  // ---- Stage 3: cooperative, aligned, coalesced b128 streaming stores ----
  // Output rows 2*row and 2*row+1 are contiguous: 3072 floats = 768 x v4f.
  v4f* __restrict__ og = (v4f*)(out + (size_t)row * (2 * OUT_ROW_F));
  const v4f* sp = (const v4f*)sout;
#pragma unroll
  for (int k = 0; k < 3; ++k) {
    __builtin_nontemporal_store(sp[k * 256 + j], &og[k * 256 + j]);
  }
}

extern "C" void kernel_launch(void* const* d_in, const int* in_sizes, int n_in,
                              void* d_out, int out_size, void* d_ws, size_t ws_size,
                              hipStream_t stream) {
  const float* x   = (const float*)d_in[0];
  float*       out = (float*)d_out;

  const int nrows = in_sizes[0] / ROW_F;    // = 32 * 256 = 8192
  subpixel_reshuffle_kernel<<<dim3(nrows), dim3(256), 0, stream>>>(x, out);
}